// GridEncoder_16999480557937
// MI455X (gfx1250) — compile-verified
//
#include <hip/hip_runtime.h>

typedef __attribute__((ext_vector_type(2))) float v2f;
typedef __attribute__((ext_vector_type(8))) float v8f;

#define GRIDN   32
#define NVOX    (GRIDN * GRIDN * GRIDN)     // 32768
#define NPTS    50000
#define NBATCH  8
#define NCH     16
#define PTS_PER_TILE 2                      // 2 points -> 16 corners -> one 16x16 WMMA tile
#define TILES_PER_BATCH (NPTS / PTS_PER_TILE)   // 25000
#define NTILES  (NBATCH * TILES_PER_BATCH)      // 200000
#define ACC_ELEMS ((size_t)NBATCH * NCH * NVOX) // 4,194,304
#define CNT_ELEMS ((size_t)NBATCH * NVOX)       // 262,144

__global__ void zero_ws(float* __restrict__ p, int n) {
    int i = blockIdx.x * blockDim.x + threadIdx.x;
    int stride = gridDim.x * blockDim.x;
    for (; i < n; i += stride) p[i] = 0.0f;
}

__global__ void encode_scatter(const float* __restrict__ x,     // [8][3][50000]
                               const float* __restrict__ W,     // [16][3]
                               const float* __restrict__ bias,  // [16]
                               float* __restrict__ acc,         // [8][16][32768]
                               float* __restrict__ cnt)         // [8][32768]
{
    const int lane        = threadIdx.x & 31;
    const int waveInBlock = threadIdx.x >> 5;
    const int wavesPerBlk = blockDim.x >> 5;
    const int waveGlobal  = blockIdx.x * wavesPerBlk + waveInBlock;
    const int nWaves      = gridDim.x * wavesPerBlk;

    const bool lowHalf = (lane < 16);
    const int  mrow    = lane & 15;   // A-matrix row held by this lane
    const int  mbase   = lowHalf ? 0 : 8;

    // A matrix (16x4 f32, K=3 padded with zeros):
    //   lanes 0-15:  a = {W[m][0], W[m][1]}   (K=0, K=1)
    //   lanes 16-31: a = {W[m][2], 0}         (K=2, K=3)
    v2f a;
    a.x = lowHalf ? W[mrow * 3 + 0] : W[mrow * 3 + 2];
    a.y = lowHalf ? W[mrow * 3 + 1] : 0.0f;

    // bias rows this lane's C/D VGPRs correspond to (M = mbase + j)
    float biasv[8];
#pragma unroll
    for (int j = 0; j < 8; ++j) biasv[j] = bias[mbase + j];

    for (int tile = waveGlobal; tile < NTILES; tile += nWaves) {
        const int batch = tile / TILES_PER_BATCH;
        const int p0    = (tile - batch * TILES_PER_BATCH) * PTS_PER_TILE;

        // Corner N handled by this lane (lanes 16-31 mirror 0-15 for B's K=2 row)
        const int N = lane & 15;
        const int p = p0 + (N >> 3);
        const int c = N & 7;            // bit2=x sel, bit1=y sel, bit0=z sel

        const float* xb = x + (size_t)batch * 3 * NPTS;
        const float rX = (xb[p]            + 0.5f) * (float)GRIDN - 0.5f;
        const float rY = (xb[NPTS + p]     + 0.5f) * (float)GRIDN - 0.5f;
        const float rZ = (xb[2 * NPTS + p] + 0.5f) * (float)GRIDN - 0.5f;

        const float fx = fminf(fmaxf(floorf(rX), 0.0f), 31.0f);
        const float cx = fminf(fmaxf(ceilf (rX), 0.0f), 31.0f);
        const float fy = fminf(fmaxf(floorf(rY), 0.0f), 31.0f);
        const float cy = fminf(fmaxf(ceilf (rY), 0.0f), 31.0f);
        const float fz = fminf(fmaxf(floorf(rZ), 0.0f), 31.0f);
        const float cz = fminf(fmaxf(ceilf (rZ), 0.0f), 31.0f);

        const float ix = (c & 4) ? cx : fx;
        const float iy = (c & 2) ? cy : fy;
        const float iz = (c & 1) ? cz : fz;

        const float rx = rX - ix, ry = rY - iy, rz = rZ - iz;
        const float d2 = rx * rx + ry * ry + rz * rz;
        const float w  = (d2 < 0.7569f) ? 1.0f : 0.0f;   // 0.87^2

        const int flat = (int)ix * (GRIDN * GRIDN) + (int)iy * GRIDN + (int)iz;

        // B matrix (4x16 f32): lanes 0-15 hold rows K=0,1; lanes 16-31 rows K=2,3(=0)
        v2f b;
        b.x = lowHalf ? rx : rz;
        b.y = lowHalf ? ry : 0.0f;

        v8f cacc = {0.f, 0.f, 0.f, 0.f, 0.f, 0.f, 0.f, 0.f};
        // D = A x B : feat[16ch][16 corners], one wave, EXEC all ones here
        v8f d = __builtin_amdgcn_wmma_f32_16x16x4_f32(
            false, a, false, b, (short)0, cacc, false, false);

        // Scatter: lane holds channels M=mbase..mbase+7 of corner N.
        // w==0 corners contribute exactly zero -> skip their atomics (~2x fewer).
        float* accb = acc + ((size_t)batch * NCH) * NVOX + flat;
        if (w != 0.0f) {
#pragma unroll
            for (int j = 0; j < 8; ++j) {
                unsafeAtomicAdd(accb + (size_t)(mbase + j) * NVOX, d[j] + biasv[j]);
            }
            if (lowHalf)
                unsafeAtomicAdd(cnt + (size_t)batch * NVOX + flat, 1.0f);
        }
    }
}

__global__ void finalize(const float* __restrict__ acc,
                         const float* __restrict__ cnt,
                         float* __restrict__ out, int total)
{
    int idx = blockIdx.x * blockDim.x + threadIdx.x;
    if (idx >= total) return;
    const int v     = idx & (NVOX - 1);
    const int bm    = idx >> 15;        // batch*16 + channel
    const int batch = bm >> 4;
    const float c   = cnt[(size_t)batch * NVOX + v];
    out[idx] = acc[idx] / fmaxf(c, 1.0f);
}

extern "C" void kernel_launch(void* const* d_in, const int* in_sizes, int n_in,
                              void* d_out, int out_size, void* d_ws, size_t ws_size,
                              hipStream_t stream) {
    (void)in_sizes; (void)n_in; (void)ws_size;
    const float* x    = (const float*)d_in[0];   // [8,3,50000]
    const float* W    = (const float*)d_in[1];   // [16,3]
    const float* bias = (const float*)d_in[2];   // [16]

    float* acc = (float*)d_ws;                   // [8][16][32768]
    float* cnt = acc + ACC_ELEMS;                // [8][32768]

    const int ztot = (int)(ACC_ELEMS + CNT_ELEMS);
    zero_ws<<<1024, 256, 0, stream>>>(acc, ztot);

    // 2048 blocks x 8 waves = 16384 waves grid-striding 200000 wave-tiles
    encode_scatter<<<2048, 256, 0, stream>>>(x, W, bias, acc, cnt);

    const int total = out_size;                  // 8*16*32768
    finalize<<<(total + 255) / 256, 256, 0, stream>>>(acc, cnt, (float*)d_out, total);
}